// FBP_sv_76484777607433
// MI455X (gfx1250) — compile-verified
//
#include <hip/hip_runtime.h>
#include <math.h>

// ---------------------------------------------------------------- constants
#define VIEWS 512
#define DETS  512
#define H_IMG 256
#define W_IMG 256

__device__ __constant__ float kD_IMG   = 0.006641f;
// VIRDET = D_DET * S2R / (S2R + D2R) = 0.0072*5.95/(5.95+4.906)
__device__ __constant__ float kS2R     = 5.95f;
__device__ __constant__ float kINV_VIRDET = (float)(1.0 / (0.0072 * 5.95 / (5.95 + 4.906)));
__device__ __constant__ float kD_ANG   = (float)(2.0 * M_PI / 512.0);

typedef __attribute__((ext_vector_type(2))) float v2f;
typedef __attribute__((ext_vector_type(8))) float v8f;

// =====================================================================
// Stage 1+2 fused: weighted projection -> ramp filter as fp32 WMMA GEMM
//   F(1024 x 512) = [proj .* w] (1024 x 512)  x  T(512 x 512)
//   T[k][d] = filt[511 + d - k]   (symmetric RL filter -> conv == corr)
// One wave computes one 16x16 tile of F, K-loop in steps of 4 using
// V_WMMA_F32_16X16X4_F32 (full fp32 precision, CDNA5 wave32 WMMA).
// Total GEMM: 1024x512x512 = 268M MACs -> negligible vs 256MB output store.
// =====================================================================
__global__ void __launch_bounds__(256)
fbp_ramp_wmma(const float* __restrict__ proj,   // (2,1,512,512) -> rows 0..1023
              const float* __restrict__ w,      // (512)
              const float* __restrict__ filt,   // (1023)
              float* __restrict__ F)            // (1024,512)
{
    const int ROWS    = 2 * VIEWS;            // 1024
    const int TILES_N = DETS / 16;            // 32

    int gwave = (blockIdx.x * blockDim.x + threadIdx.x) >> 5;
    int lane  = threadIdx.x & 31;
    int tm    = gwave / TILES_N;              // 0..63   (row tile)
    int tn    = gwave % TILES_N;              // 0..31   (col tile)
    int rowBase = tm * 16;
    int colBase = tn * 16;
    if (rowBase >= ROWS) return;

    int l16  = lane & 15;
    int half = lane >> 4;                     // 0: K pair {k0,k0+1}; 1: {k0+2,k0+3}

    const float* prow = proj + (size_t)(rowBase + l16) * DETS;  // A row for this lane
    const int    n    = colBase + l16;                          // B column for this lane

    v8f acc = {};

    #pragma unroll 8
    for (int k0 = 0; k0 < DETS; k0 += 4) {
        int ka = k0 + half * 2;
        v2f a, b;
        // A fragment: q[m][k] = proj[m][k] * w[k]   (fused detector weighting)
        a.x = prow[ka]     * w[ka];
        a.y = prow[ka + 1] * w[ka + 1];
        // B fragment: T[k][n] = filt[511 + n - k]   (Toeplitz, read in place)
        b.x = filt[(DETS - 1) + n - ka];
        b.y = filt[(DETS - 1) + n - ka - 1];
        acc = __builtin_amdgcn_wmma_f32_16x16x4_f32(
                  /*neg_a=*/false, a, /*neg_b=*/false, b,
                  /*c_mod=*/(short)0, acc,
                  /*reuse_a=*/false, /*reuse_b=*/false);
    }

    // C/D layout: VGPR r holds row M = r + 8*half, col N = l16
    float* Fo = F + colBase + l16;
    #pragma unroll
    for (int r = 0; r < 8; ++r)
        Fo[(size_t)(rowBase + r + 8 * half) * DETS] = acc[r];
}

// =====================================================================
// Stage 3: per-view backprojection (no view reduction; output (2,512,256,256))
// grid.x = 1024 (b*512+v), grid.y = 16 row tiles of 16 image rows.
// The 512-float filtered row lives in LDS (ds_load gathers); stores are
// coalesced float4. Per-pixel ALU trimmed (~15 VALU + 1 v_rcp + 2 ds) so
// the kernel stays pinned to the output-store roofline:
// 256 MB @ 23.3 TB/s -> ~11 us floor on MI455X.
// =====================================================================
__global__ void __launch_bounds__(256)
fbp_backproject(const float* __restrict__ F,    // (1024,512)
                float* __restrict__ out)        // (2,512,256,256)
{
    __shared__ float frow[DETS];

    const int row = blockIdx.x;               // b*512 + v
    const int v   = row & (VIEWS - 1);

    // stage this view's filtered row into LDS
    for (int i = threadIdx.x; i < DETS; i += blockDim.x)
        frow[i] = F[(size_t)row * DETS + i];
    __syncthreads();

    float beta = kD_ANG * (float)v;           // ANG0 == 0
    float sb, cb;
    sincosf(beta, &sb, &cb);

    const int tid  = threadIdx.x;
    const int ww   = (tid & 63) * 4;          // 4 consecutive pixels along W
    const int hsub = tid >> 6;                // 0..3
    const int h0   = blockIdx.y * 16;

    const float s2r2 = kS2R * kS2R;
    float4* out4 = (float4*)(out + (size_t)row * H_IMG * W_IMG);

    #pragma unroll
    for (int r = 0; r < 4; ++r) {
        int   h   = h0 + hsub * 4 + r;
        float y   = (127.5f - (float)h) * kD_IMG;
        // d(x)   = (S2R - y*sb) - x*cb          (1 FMA per pixel)
        // num(x) = S2R*y*cb     - x*(S2R*sb)    (1 FMA per pixel)
        float d0  = kS2R - y * sb;
        float n0  = kS2R * y * cb;
        float nsb = kS2R * sb;

        float4 o;
        #pragma unroll
        for (int j = 0; j < 4; ++j) {
            float x   = ((float)(ww + j) - 127.5f) * kD_IMG;
            float d   = d0 - x * cb;
            float num = n0 - x * nsb;
            float rd  = __builtin_amdgcn_rcpf(d);       // v_rcp_f32, ~1 ulp
            float t   = num * rd * kINV_VIRDET + 255.5f;
            float wg  = s2r2 * rd * rd;

            float tf   = floorf(t);
            int   i0   = (int)tf;
            float frac = t - tf;
            // unsigned-compare masks; gather index wrapped into [0,511]
            // (out-of-range taps are zeroed by the mask, index just must
            //  stay inside the LDS array)
            float m0 = ((unsigned)i0       < (unsigned)DETS) ? 1.0f : 0.0f;
            float m1 = ((unsigned)(i0 + 1) < (unsigned)DETS) ? 1.0f : 0.0f;
            int   i0c = i0 & (DETS - 1);
            int   i1c = (i0 + 1) & (DETS - 1);

            float f0 = frow[i0c];
            float f1 = frow[i1c];
            float val = wg * (f0 * ((1.0f - frac) * m0) + f1 * (frac * m1));
            (&o.x)[j] = val;
        }
        out4[(size_t)h * (W_IMG / 4) + (ww >> 2)] = o;
    }
}

// ---------------------------------------------------------------- launcher
extern "C" void kernel_launch(void* const* d_in, const int* in_sizes, int n_in,
                              void* d_out, int out_size, void* d_ws, size_t ws_size,
                              hipStream_t stream) {
    const float* proj = (const float*)d_in[0];  // (2,1,512,512)
    const float* w    = (const float*)d_in[1];  // (512)
    const float* filt = (const float*)d_in[2];  // (1023)
    float*       out  = (float*)d_out;          // (2,512,256,256)
    float*       F    = (float*)d_ws;           // 1024*512 f32 = 2 MB scratch

    // GEMM: 64x32 = 2048 wave-tiles, 8 waves/block -> 256 blocks
    fbp_ramp_wmma<<<256, 256, 0, stream>>>(proj, w, filt, F);

    // Backprojection: one block per (b,v) x 16-row tile
    fbp_backproject<<<dim3(2 * VIEWS, H_IMG / 16), 256, 0, stream>>>(F, out);
}